// SSIM_4277787426993
// MI455X (gfx1250) — compile-verified
//
#include <hip/hip_runtime.h>

typedef __attribute__((ext_vector_type(2))) float v2f;
typedef __attribute__((ext_vector_type(8))) float v8f;

#define IMG_H 512
#define IMG_W 512
#define NPLANES 48                 // 16 * 3
#define TOTAL_TILES 49152          // 48 * 32 * 32 (16x16 tiles)
#define WAVES_PER_BLOCK 8
#define NBLOCKS (TOTAL_TILES / WAVES_PER_BLOCK)   // 6144
#define LDS_PER_WAVE 1904          // 728 (x) + 728 (y) + 448 (T col-major 16x28)

// Compile-time per-lane Toeplitz operand table for the 11-tap Gaussian
// (sigma=1.5). Shared by horizontal-B and vertical-A WMMA operands:
//   tab[lane][2q+j] = w[(4q + 2*(lane>>4) + j) - (lane&15)]  (0 outside [0,10])
struct alignas(16) TwTab { float v[32][14]; };

static constexpr TwTab make_twtab() {
  TwTab t{};
  constexpr float Wt[11] = {0.0010283846f, 0.0075987582f, 0.0360008030f,
                            0.1093606778f, 0.2130055329f, 0.2660116899f,
                            0.2130055329f, 0.1093606778f, 0.0360008030f,
                            0.0075987582f, 0.0010283846f};
  for (int lane = 0; lane < 32; ++lane) {
    const int n16 = lane & 15;
    const int kh  = lane >> 4;
    for (int q = 0; q < 7; ++q)
      for (int j = 0; j < 2; ++j) {
        const int d = 4 * q + 2 * kh + j - n16;
        t.v[lane][2 * q + j] = (d >= 0 && d <= 10) ? Wt[d] : 0.0f;
      }
  }
  return t;
}

__device__ constexpr TwTab g_twtab = make_twtab();

__global__ __launch_bounds__(256) void ssim_wmma_kernel(
    const float* __restrict__ x, const float* __restrict__ y,
    float* __restrict__ blocksums) {
  __shared__ float smem[WAVES_PER_BLOCK * LDS_PER_WAVE];
  __shared__ float wavesum[WAVES_PER_BLOCK];

  const int lane = threadIdx.x & 31;
  const int wave = threadIdx.x >> 5;
  const int n16  = lane & 15;   // A-matrix M index / B-matrix N index (ISA layout)
  const int kh   = lane >> 4;   // K half selector (ISA layout)

  float* sxl = smem + wave * LDS_PER_WAVE;  // 26x28 halo tile of x
  float* syl = sxl + 728;                   // 26x28 halo tile of y
  float* sT  = sxl + 1456;                  // intermediate T, column-major [16][28]

  const int tile  = blockIdx.x * WAVES_PER_BLOCK + wave;
  const int plane = tile >> 10;
  const int trow  = (tile >> 5) & 31;
  const int tcol  = tile & 31;
  const int R0 = trow * 16;
  const int C0 = tcol * 16;
  const int planeOff = plane * (IMG_H * IMG_W);

  // zero T pad rows 26,27 (read by last vertical K-chunk; Toeplitz weight is 0,
  // but 0*NaN = NaN inside WMMA so they must be finite)
  sT[n16 * 28 + 26 + kh] = 0.0f;

  // stage 26x28 halo tiles (zero-padded at image borders = conv zero padding)
  for (int i = lane; i < 728; i += 32) {
    int r = i / 28;
    int c = i - r * 28;
    int gr = R0 - 5 + r;
    int gc = C0 - 5 + c;
    float vx = 0.0f, vy = 0.0f;
    if ((unsigned)gr < (unsigned)IMG_H && (unsigned)gc < (unsigned)IMG_W) {
      int g = planeOff + gr * IMG_W + gc;
      vx = x[g];
      vy = y[g];
    }
    sxl[i] = vx;
    syl[i] = vy;
  }

  // per-lane Toeplitz operands from the constexpr table (vectorized loads)
  v2f tw[7];
#pragma unroll
  for (int q = 0; q < 7; ++q)
    tw[q] = *(const v2f*)&g_twtab.v[lane][2 * q];

  // Intra-wave LDS producer->consumer only: DS ops from one wave execute
  // in order, so no block barrier is needed -- just stop compiler motion.
  __builtin_amdgcn_wave_barrier();

  v8f res[5];
#pragma unroll
  for (int ch = 0; ch < 5; ++ch) {
    // ---- horizontal pass: two M-tiles (rows 0..15 and 10..25) ----
    v8f d1 = {};
    v8f d2 = {};
#pragma unroll
    for (int q = 0; q < 7; ++q) {
      const int cb = 4 * q + 2 * kh;
      {
        v2f ax = *(const v2f*)&sxl[n16 * 28 + cb];
        v2f ay = *(const v2f*)&syl[n16 * 28 + cb];
        v2f a;
        if (ch == 0) a = ax;
        else if (ch == 1) a = ay;
        else if (ch == 2) a = ax * ax;
        else if (ch == 3) a = ay * ay;
        else a = ax * ay;
        d1 = __builtin_amdgcn_wmma_f32_16x16x4_f32(false, a, false, tw[q],
                                                   (short)0, d1, false, false);
      }
      {
        v2f ax = *(const v2f*)&sxl[(n16 + 10) * 28 + cb];
        v2f ay = *(const v2f*)&syl[(n16 + 10) * 28 + cb];
        v2f a;
        if (ch == 0) a = ax;
        else if (ch == 1) a = ay;
        else if (ch == 2) a = ax * ax;
        else if (ch == 3) a = ay * ay;
        else a = ax * ay;
        d2 = __builtin_amdgcn_wmma_f32_16x16x4_f32(false, a, false, tw[q],
                                                   (short)0, d2, false, false);
      }
    }
    // scatter T to LDS column-major: D[p] = T[p + 8*kh][n16].
    // Tile-2 rows 10..15 are bitwise identical to tile-1's (same A rows, same
    // tw, same chunk order), so store d2 unconditionally -- no EXEC branching;
    // same-wave DS ops are in-order so the benign overwrite is race-free.
#pragma unroll
    for (int p = 0; p < 8; ++p) sT[n16 * 28 + p + 8 * kh] = d1[p];
#pragma unroll
    for (int p = 0; p < 8; ++p) sT[n16 * 28 + 10 + p + 8 * kh] = d2[p];
    __builtin_amdgcn_wave_barrier();      // same-wave DS in-order: RAW safe

    // ---- vertical pass: Out = Toeplitz(16x28) x T(28x16) ----
    v8f acc = {};
#pragma unroll
    for (int q = 0; q < 7; ++q) {
      v2f b = *(const v2f*)&sT[n16 * 28 + 4 * q + 2 * kh];
      acc = __builtin_amdgcn_wmma_f32_16x16x4_f32(false, tw[q], false, b,
                                                  (short)0, acc, false, false);
    }
    res[ch] = acc;
    __builtin_amdgcn_wave_barrier();      // WAR: next channel rewrites T
  }

  // ---- SSIM map + reduction ----
  const float C1 = 1e-4f;   // 0.01^2
  const float C2 = 9e-4f;   // 0.03^2
  float local = 0.0f;
#pragma unroll
  for (int p = 0; p < 8; ++p) {
    float mx = res[0][p], my = res[1][p];
    float mxs = mx * mx, mys = my * my, mxy = mx * my;
    float sgx = res[2][p] - mxs;
    float sgy = res[3][p] - mys;
    float sgxy = res[4][p] - mxy;
    float num = (2.0f * mxy + C1) * (2.0f * sgxy + C2);
    float den = (mxs + mys + C1) * (sgx + sgy + C2);
    local += num / (den + 1e-8f);
  }
#pragma unroll
  for (int off = 16; off > 0; off >>= 1) local += __shfl_xor(local, off, 32);
  if (lane == 0) wavesum[wave] = local;
  __syncthreads();  // only cross-wave sync in the kernel
  if (threadIdx.x == 0) {
    float s = 0.0f;
    for (int w2 = 0; w2 < WAVES_PER_BLOCK; ++w2) s += wavesum[w2];
    blocksums[blockIdx.x] = s;
  }
}

__global__ __launch_bounds__(256) void ssim_reduce_kernel(
    const float* __restrict__ bs, float* __restrict__ out, int n) {
  __shared__ double sd[256];
  double s = 0.0;
  for (int i = threadIdx.x; i < n; i += 256) s += (double)bs[i];
  sd[threadIdx.x] = s;
  __syncthreads();
  for (int off = 128; off > 0; off >>= 1) {
    if ((int)threadIdx.x < off) sd[threadIdx.x] += sd[threadIdx.x + off];
    __syncthreads();
  }
  if (threadIdx.x == 0)
    out[0] = (float)(sd[0] * (1.0 / (double)(NPLANES * IMG_H * IMG_W)));
}

extern "C" void kernel_launch(void* const* d_in, const int* in_sizes, int n_in,
                              void* d_out, int out_size, void* d_ws, size_t ws_size,
                              hipStream_t stream) {
  (void)in_sizes; (void)n_in; (void)out_size; (void)ws_size;
  const float* x = (const float*)d_in[0];
  const float* y = (const float*)d_in[1];
  float* blocksums = (float*)d_ws;  // NBLOCKS floats = 24 KB
  ssim_wmma_kernel<<<NBLOCKS, 256, 0, stream>>>(x, y, blocksums);
  ssim_reduce_kernel<<<1, 256, 0, stream>>>(blocksums, (float*)d_out, NBLOCKS);
}